// GCA_27556510171451
// MI455X (gfx1250) — compile-verified
//
#include <hip/hip_runtime.h>
#include <hip/hip_bf16.h>
#include <math.h>

// Problem constants (from reference):
//   N=16, D=512, H=48, W=160, HEADS=8 -> B = N*HEADS = 128, S = 64, P = H*W = 7680
//   NECK = 128
#define GC_B    128
#define GC_S    64
#define GC_P    7680
#define GC_PC   (GC_P / 4)      // 1920 float4 chunks per row
#define GC_N    16
#define GC_D    512
#define GC_NECK 128

typedef float v2f __attribute__((ext_vector_type(2)));
typedef float v8f __attribute__((ext_vector_type(8)));

// ---------------------------------------------------------------------------
// Kernel 1: per-(n*head) softmax-weighted global pooling.
// grid = 128 blocks (one per b), block = 1024 threads (32 wave32 waves).
// ---------------------------------------------------------------------------
__global__ __launch_bounds__(1024, 1)
void gca_pool_kernel(const float* __restrict__ x,
                     const float* __restrict__ mask_w,
                     const float* __restrict__ mask_b,
                     float* __restrict__ y /* (128,64) */) {
    __shared__ __align__(16) float l_attn[GC_P];   // 30 KB: logits -> exp weights
    __shared__ float l_mw[GC_S];
    __shared__ float l_red[32];

    const int b    = blockIdx.x;
    const int t    = threadIdx.x;
    const int lane = t & 31;
    const int wv   = t >> 5;            // 0..31
    const float* __restrict__ xb = x + (size_t)b * GC_S * GC_P;

    if (t < GC_S) l_mw[t] = mask_w[t];
    __syncthreads();

    const float mb = mask_b[0];

    // ---- Pass A: logits[p] = sum_s x[s,p] * mw[s] + mb, float4-vectorized ----
    float lmax = -3.402823466e38f;
    for (int ch = t; ch < GC_PC; ch += 1024) {
        float ax = 0.f, ay = 0.f, az = 0.f, aw = 0.f;
        const float* col = xb + ch * 4;
        #pragma unroll 4
        for (int s = 0; s < GC_S; ++s) {
            const float w  = l_mw[s];
            const float4 v = *(const float4*)(col + (size_t)s * GC_P);
            ax = fmaf(v.x, w, ax); ay = fmaf(v.y, w, ay);
            az = fmaf(v.z, w, az); aw = fmaf(v.w, w, aw);
        }
        float4 lg = make_float4(ax + mb, ay + mb, az + mb, aw + mb);
        ((float4*)l_attn)[ch] = lg;
        lmax = fmaxf(lmax, fmaxf(fmaxf(lg.x, lg.y), fmaxf(lg.z, lg.w)));
    }
    __syncthreads();

    // ---- block max reduction (wave shfl + 32 LDS partials) ----
    #pragma unroll
    for (int o = 16; o; o >>= 1) lmax = fmaxf(lmax, __shfl_xor(lmax, o));
    if (lane == 0) l_red[wv] = lmax;
    __syncthreads();
    float gmax = l_red[0];
    #pragma unroll
    for (int i = 1; i < 32; ++i) gmax = fmaxf(gmax, l_red[i]);
    __syncthreads();

    // ---- exp in place + sum ----
    float lsum = 0.f;
    for (int ch = t; ch < GC_PC; ch += 1024) {
        float4 lg = ((float4*)l_attn)[ch];
        lg.x = __expf(lg.x - gmax); lg.y = __expf(lg.y - gmax);
        lg.z = __expf(lg.z - gmax); lg.w = __expf(lg.w - gmax);
        ((float4*)l_attn)[ch] = lg;
        lsum += lg.x + lg.y + lg.z + lg.w;
    }
    __syncthreads();
    #pragma unroll
    for (int o = 16; o; o >>= 1) lsum += __shfl_xor(lsum, o);
    if (lane == 0) l_red[wv] = lsum;
    __syncthreads();
    float gsum = 0.f;
    #pragma unroll
    for (int i = 0; i < 32; ++i) gsum += l_red[i];
    __syncthreads();
    const float inv = 1.0f / gsum;

    // ---- Pass B: y[s] = (sum_p x[s,p] * e[p]) * inv; 2 rows per wave ----
    for (int s = wv; s < GC_S; s += 32) {
        const float* row = xb + (size_t)s * GC_P;
        float acc = 0.f;
        for (int ch = lane; ch < GC_PC; ch += 32) {
            const float4 v = *(const float4*)(row + ch * 4);
            const float4 a = ((const float4*)l_attn)[ch];
            acc = fmaf(v.x, a.x, acc); acc = fmaf(v.y, a.y, acc);
            acc = fmaf(v.z, a.z, acc); acc = fmaf(v.w, a.w, acc);
        }
        #pragma unroll
        for (int o = 16; o; o >>= 1) acc += __shfl_xor(acc, o);
        if (lane == 0) y[b * GC_S + s] = acc * inv;
    }
}

// ---------------------------------------------------------------------------
// Kernel 2: bottleneck MLP via fp32 WMMA (V_WMMA_F32_16X16X4_F32).
//   z  = y(16x512) @ w1^T(512x128) + b1 ; LayerNorm(row) ; ReLU
//   z2 = z(16x128) @ w2^T(128x512) + b2
// one block, 256 threads = 8 waves. EXEC all-1s around every WMMA.
// ---------------------------------------------------------------------------
__global__ __launch_bounds__(256, 1)
void gca_mlp_kernel(const float* __restrict__ y,
                    const float* __restrict__ w1, const float* __restrict__ b1,
                    const float* __restrict__ ln_w, const float* __restrict__ ln_b,
                    const float* __restrict__ w2, const float* __restrict__ b2,
                    float* __restrict__ z2 /* (16,512) */) {
    __shared__ float zl[GC_N * GC_NECK];          // 16 x 128

    const int t    = threadIdx.x;
    const int lane = t & 31;
    const int wv   = t >> 5;       // 0..7
    const int half = lane >> 4;    // 0: K pair {0,1} / M rows 0..7 ; 1: K pair {2,3} / M rows 8..15
    const int lid  = lane & 15;

    // ---------------- GEMM1: M=16, N=128, K=512; wave wv owns N-tile wv*16 --
    {
        const int n0 = wv * 16;
        const int nc = n0 + lid;                  // output column handled by lane
        v8f acc = {};
        for (int k0 = 0; k0 < GC_D; k0 += 4) {
            const int ka = k0 + half * 2;
            // A(16x4) per ISA 32-bit A layout: lane<16 -> K=0,1 ; lane>=16 -> K=2,3 ; M = lid
            v2f a;
            a.x = y[lid * GC_D + ka];
            a.y = y[lid * GC_D + ka + 1];
            // B(4x16) = w1^T : B[k][n] = w1[n*512 + k]; mirrored lane layout
            v2f bb;
            bb.x = w1[(size_t)nc * GC_D + ka];
            bb.y = w1[(size_t)nc * GC_D + ka + 1];
            acc = __builtin_amdgcn_wmma_f32_16x16x4_f32(
                false, a, false, bb, (short)0, acc, false, false);
        }
        // C/D layout: lane<16 -> rows 0..7, N=lane ; lane>=16 -> rows 8..15, N=lane-16
        const float bias = b1[nc];
        #pragma unroll
        for (int v = 0; v < 8; ++v)
            zl[(v + half * 8) * GC_NECK + nc] = acc[v] + bias;
    }
    __syncthreads();

    // ---------------- LayerNorm(128) + ReLU per row (16 rows) ---------------
    if (t < GC_N) {
        float s1 = 0.f, s2 = 0.f;
        for (int c = 0; c < GC_NECK; ++c) {
            const float v = zl[t * GC_NECK + c];
            s1 += v; s2 += v * v;
        }
        const float mu  = s1 * (1.0f / GC_NECK);
        const float var = s2 * (1.0f / GC_NECK) - mu * mu;
        const float rs  = rsqrtf(var + 1e-5f);
        for (int c = 0; c < GC_NECK; ++c) {
            const float v = (zl[t * GC_NECK + c] - mu) * rs * ln_w[c] + ln_b[c];
            zl[t * GC_NECK + c] = fmaxf(v, 0.f);
        }
    }
    __syncthreads();

    // ---------------- GEMM2: M=16, N=512, K=128; wave wv owns 4 N-tiles -----
    for (int tile = 0; tile < 4; ++tile) {
        const int n0 = wv * 64 + tile * 16;
        const int nc = n0 + lid;
        v8f acc = {};
        for (int k0 = 0; k0 < GC_NECK; k0 += 4) {
            const int ka = k0 + half * 2;
            v2f a;
            a.x = zl[lid * GC_NECK + ka];
            a.y = zl[lid * GC_NECK + ka + 1];
            v2f bb;                                   // B[k][n] = w2[n*128 + k]
            bb.x = w2[(size_t)nc * GC_NECK + ka];
            bb.y = w2[(size_t)nc * GC_NECK + ka + 1];
            acc = __builtin_amdgcn_wmma_f32_16x16x4_f32(
                false, a, false, bb, (short)0, acc, false, false);
        }
        const float bias = b2[nc];
        #pragma unroll
        for (int v = 0; v < 8; ++v)
            z2[(size_t)(v + half * 8) * GC_D + nc] = acc[v] + bias;
    }
}

// ---------------------------------------------------------------------------
// Kernel 3: out[n,c,h,w] = x[n,c,h,w] + z2[n,c]   (pure bandwidth)
// one block per (n,c) row: scalar broadcast + float4 streaming.
// ---------------------------------------------------------------------------
__global__ __launch_bounds__(256, 4)
void gca_add_kernel(const float* __restrict__ x,
                    const float* __restrict__ z2,
                    float* __restrict__ out) {
    const int row = blockIdx.x;                 // 0 .. 16*512-1
    const float add = z2[row];
    const float4* __restrict__ xr = (const float4*)(x   + (size_t)row * GC_P);
    float4*       __restrict__ orow = (float4*)(out + (size_t)row * GC_P);
    for (int ch = threadIdx.x; ch < GC_PC; ch += 256) {
        float4 v = xr[ch];
        v.x += add; v.y += add; v.z += add; v.w += add;
        orow[ch] = v;
    }
}

// ---------------------------------------------------------------------------
extern "C" void kernel_launch(void* const* d_in, const int* in_sizes, int n_in,
                              void* d_out, int out_size, void* d_ws, size_t ws_size,
                              hipStream_t stream) {
    const float* x      = (const float*)d_in[0];
    const float* mask_w = (const float*)d_in[1];
    const float* mask_b = (const float*)d_in[2];
    const float* w1     = (const float*)d_in[3];
    const float* b1     = (const float*)d_in[4];
    const float* ln_w   = (const float*)d_in[5];
    const float* ln_b   = (const float*)d_in[6];
    const float* w2     = (const float*)d_in[7];
    const float* b2     = (const float*)d_in[8];
    float* out = (float*)d_out;

    float* y  = (float*)d_ws;          // (128,64)  pooled context
    float* z2 = y + GC_B * GC_S;       // (16,512)  MLP output

    gca_pool_kernel<<<GC_B, 1024, 0, stream>>>(x, mask_w, mask_b, y);
    gca_mlp_kernel<<<1, 256, 0, stream>>>(y, w1, b1, ln_w, ln_b, w2, b2, z2);
    gca_add_kernel<<<GC_N * GC_D, 256, 0, stream>>>(x, z2, out);
}